// feat_gan_47467978555823
// MI455X (gfx1250) — compile-verified
//
#include <hip/hip_runtime.h>

typedef __attribute__((ext_vector_type(2))) float v2f;
typedef __attribute__((ext_vector_type(8))) float v8f;

#define BALL_EPS 0.01f
#define IDX_INF  0x7FFFFFFF

// ---- wave32 cross-lane helpers via LDS permute hardware (no LDS memory) ----
static __device__ __forceinline__ int bperm_i(int src_lane, int v) {
  return __builtin_amdgcn_ds_bpermute(src_lane << 2, v);
}
static __device__ __forceinline__ float bperm_f(int src_lane, float v) {
  return __int_as_float(__builtin_amdgcn_ds_bpermute(src_lane << 2, __float_as_int(v)));
}

// Exact reference-ordered f32 distance^2: ((dx*dx + dy*dy) + dz*dz).
// Contraction disabled so no fmac sneaks in: must match XLA's
// elementwise-square + add ordering bit-for-bit at the d2 < 1 threshold.
static __device__ __forceinline__ float exact_d2(float qx, float qy, float qz,
                                                 float sx, float sy, float sz) {
#pragma clang fp contract(off)
  float dx = qx - sx, dy = qy - sy, dz = qz - sz;
  return (dx * dx + dy * dy) + dz * dz;
}

// Per query column (16 queries per wave, duplicated across lane halves):
// smallest source index j with ||q - src[j]||^2 < 1, else IDX_INF.
//
// WMMA prefilter: d2 ~= (|s|^2 - 2 q.s) + |q|^2 via TWO chained
// V_WMMA_F32_16X16X4_F32 ops that place data only in the K0 / K3 slots
// (v0/lanes0-15 and v1/lanes16-31), which map identically under either
// plausible B-fragment layout convention; ambiguous slots carry zeros.
// Candidates inside the 1+EPS band are re-verified with exact arithmetic.
static __device__ __forceinline__ int ball_scan_first(
    const float* __restrict__ src, int M,
    float qx, float qy, float qz, float qn, v2f bfrag1, v2f bfrag2,
    int lane, int col, int hi) {
  int first = IDX_INF;
  const int ntiles = M >> 4;
  for (int jt = 0; jt < ntiles; ++jt) {
    const float* sp = src + 3 * ((jt << 4) + col);
    float sx = sp[0], sy = sp[1], sz = sp[2];
    if (jt + 1 < ntiles)
      __builtin_prefetch(src + 3 * (((jt + 1) << 4) + col), 0, 1);

    float sn = (sx * sx + sy * sy) + sz * sz;   // prefilter-only, fma OK
    // A fragments (sources, 16x4 f32): only K0 (lanes0-15,v0) and
    // K3 (lanes16-31,v1) carry data; K1/K2 slots are zero.
    v2f a1, a2;
    if (hi) { a1.x = 0.0f; a1.y = sy; a2.x = 0.0f; a2.y = sn; }
    else    { a1.x = sx;   a1.y = 0.0f; a2.x = sz; a2.y = 0.0f; }

    v8f c0 = {};
    // D1[m][n] = -2(sx*qx + sy*qy)
    v8f d1 = __builtin_amdgcn_wmma_f32_16x16x4_f32(false, a1, false, bfrag1,
                                                   (short)0, c0, false, false);
    // D[m][n] = |s|^2 - 2 s.q
    v8f d = __builtin_amdgcn_wmma_f32_16x16x4_f32(false, a2, false, bfrag2,
                                                  (short)0, d1, false, false);

    int lf = IDX_INF;
    if (first == IDX_INF) {          // pair-uniform branch (first is pair-synced)
      const int base = (jt << 4) + (hi << 3);   // D row m = r + 8*hi
#pragma unroll
      for (int r = 0; r < 8; ++r) {
        float approx = d[r] + qn;
        if (approx < 1.0f + BALL_EPS) {
          const float* tp = src + 3 * (base + r);
          if (exact_d2(qx, qy, qz, tp[0], tp[1], tp[2]) < 1.0f) {
            lf = base + r;
            break;                    // rows ascend -> first hit is minimal
          }
        }
      }
    }
    // combine lane pair (l, l^16): full-EXEC permute, min keeps earliest index
    int other = bperm_i(lane ^ 16, lf);
    int cand = lf < other ? lf : other;
    first = cand < first ? cand : first;

    if (!__any(first == IDX_INF)) break;   // wave-uniform early exit
  }
  return first;
}

__global__ __launch_bounds__(32) void feat_gan_layer_kernel(
    const float* __restrict__ att_xyz, const float* __restrict__ bat_xyz,
    const float* __restrict__ att_feat, const float* __restrict__ bat_feat,
    int N, int C, double* __restrict__ acc) {
  const int lane = threadIdx.x;
  const int col = lane & 15;
  const int hi = lane >> 4;
  const int tiles_per_b = N >> 4;
  const int b = blockIdx.x / tiles_per_b;
  const int qt = blockIdx.x % tiles_per_b;
  const int qi = (qt << 4) + col;

  const float* bx = bat_xyz + (size_t)b * N * 3;
  const float* ax = att_xyz + (size_t)b * N * 3;
  const float* afeat = att_feat + (size_t)b * C * N;
  const float* bfeat = bat_feat + (size_t)b * C * N;

  const float qx = bx[3 * qi + 0];
  const float qy = bx[3 * qi + 1];
  const float qz = bx[3 * qi + 2];
  const float qn = (qx * qx + qy * qy) + qz * qz;   // prefilter-only

  // B fragments (queries, 4x16 f32): data only in K0 / K3 slots.
  //   bfrag1: K0=-2qx, K3=-2qy ; bfrag2: K0=-2qz, K3=1
  v2f bfrag1, bfrag2;
  if (hi) { bfrag1.x = 0.0f;       bfrag1.y = -2.0f * qy;
            bfrag2.x = 0.0f;       bfrag2.y = 1.0f; }
  else    { bfrag1.x = -2.0f * qx; bfrag1.y = 0.0f;
            bfrag2.x = -2.0f * qz; bfrag2.y = 0.0f; }

  const int bat_first =
      ball_scan_first(bx, N, qx, qy, qz, qn, bfrag1, bfrag2, lane, col, hi);
  const int att_first =
      ball_scan_first(ax, N, qx, qy, qz, qn, bfrag1, bfrag2, lane, col, hi);

  // Masked squared-error for this query; lane pair splits the 3+C channels.
  float partial = 0.0f;
  if (att_first != IDX_INF) {
    const int ai = att_first;
    const int bi = bat_first;          // always valid (self is within radius)
    const int CT = C + 3;
    for (int ch = hi; ch < CT; ch += 2) {
      float va, vb;
      if (ch < 3) { va = ax[3 * ai + ch];                  vb = bx[3 * bi + ch]; }
      else        { va = afeat[(size_t)(ch - 3) * N + ai]; vb = bfeat[(size_t)(ch - 3) * N + bi]; }
      float dd = va - vb;
      partial += dd * dd;
    }
  }
  // wave32 butterfly reduction
  for (int off = 16; off; off >>= 1)
    partial += bperm_f(lane ^ off, partial);
  if (lane == 0) atomicAdd(acc, (double)partial);
}

__global__ void feat_gan_init_kernel(double* __restrict__ acc) {
  acc[0] = 0.0;
  acc[1] = 0.0;
}

__global__ void feat_gan_final_kernel(const double* __restrict__ acc,
                                      float* __restrict__ out,
                                      double inv0, double inv1) {
  float l0 = (float)(acc[0] * inv0);
  float l1 = (float)(acc[1] * inv1);
  float l = 0.5f * (l0 + l1);
  out[0] = (l != l) ? l1 : l;   // jnp.where(isnan(loss), losses[-1], loss)
}

extern "C" void kernel_launch(void* const* d_in, const int* in_sizes, int n_in,
                              void* d_out, int out_size, void* d_ws, size_t ws_size,
                              hipStream_t stream) {
  const float* att_xyz0  = (const float*)d_in[0];
  const float* att_xyz1  = (const float*)d_in[1];
  const float* bat_xyz0  = (const float*)d_in[2];
  const float* bat_xyz1  = (const float*)d_in[3];
  const float* att_feat0 = (const float*)d_in[4];
  const float* att_feat1 = (const float*)d_in[5];
  const float* bat_feat0 = (const float*)d_in[6];
  const float* bat_feat1 = (const float*)d_in[7];

  const int B  = 4;
  const int N0 = in_sizes[0] / (B * 3);
  const int N1 = in_sizes[1] / (B * 3);
  const int C0 = in_sizes[4] / (B * N0);
  const int C1 = in_sizes[5] / (B * N1);

  double* acc = (double*)d_ws;
  feat_gan_init_kernel<<<1, 1, 0, stream>>>(acc);
  feat_gan_layer_kernel<<<B * (N0 >> 4), 32, 0, stream>>>(
      att_xyz0, bat_xyz0, att_feat0, bat_feat0, N0, C0, acc + 0);
  feat_gan_layer_kernel<<<B * (N1 >> 4), 32, 0, stream>>>(
      att_xyz1, bat_xyz1, att_feat1, bat_feat1, N1, C1, acc + 1);
  feat_gan_final_kernel<<<1, 1, 0, stream>>>(
      acc, (float*)d_out,
      1.0 / ((double)B * N0 * (3 + C0)),
      1.0 / ((double)B * N1 * (3 + C1)));
}